// PointBasedTransform_29403346108658
// MI455X (gfx1250) — compile-verified
//
#include <hip/hip_runtime.h>
#include <stdint.h>

// ---------------- problem constants (match reference) ----------------
static constexpr int   GRID_SIZE = 64;
static constexpr float PITCH     = 0.02f;
static constexpr float RESF      = 0.04f;
static constexpr int   MAX_PTS   = 16;
static constexpr int   MIN_PTS   = 15;
static constexpr int   G         = 32;
static constexpr int   NUM_VOX   = G * G * G;        // 32768
static constexpr int   NPTS      = 8 * 384 * 512;    // 1572864
static constexpr int   NBUCK     = 65536;
static constexpr float HALF_CUBE = GRID_SIZE * PITCH * 0.5f;  // 0.64

// ---------------- output layout (floats, concatenated in return order) ----
static constexpr long long O_PTS   = 0;
static constexpr long long O_CONF  = O_PTS   + 3LL * NPTS;
static constexpr long long O_DESC  = O_CONF  + NPTS;
static constexpr long long O_DCONF = O_DESC  + 24LL * NPTS;
static constexpr long long O_TABLE = O_DCONF + NPTS;
static constexpr long long O_OCC   = O_TABLE + (long long)NUM_VOX * MAX_PTS;
static constexpr long long O_IMG   = O_OCC   + NUM_VOX;
static constexpr long long O_GRID  = O_IMG   + NPTS;
static constexpr long long O_TOTAL = O_GRID  + NUM_VOX;

// ---------------- workspace layout (4-byte elements) ----------------
static constexpr long long W_COUNTS0 = 0;                       // int[33024] (NUM_VOX+1 padded)
static constexpr long long W_FILL    = W_COUNTS0 + 33024;       // int[NUM_VOX]
static constexpr long long W_HIST    = W_FILL    + NUM_VOX;     // int[NBUCK]
static constexpr long long W_BFILL   = W_HIST    + NBUCK;       // int[NBUCK]
static constexpr long long W_ZEND    = W_BFILL   + NBUCK;       // contiguous zero region
static constexpr long long W_VID     = W_ZEND;                  // int[NPTS]
static constexpr long long W_SIDX    = W_VID   + NPTS;          // int[NPTS]
static constexpr long long W_OFF     = W_SIDX  + NPTS;          // int[NUM_VOX]
static constexpr long long W_BOFF    = W_OFF   + NUM_VOX;       // int[NBUCK]
static constexpr long long W_CNTF    = W_BOFF  + NBUCK;         // float[NUM_VOX]
static constexpr long long W_SCANV   = W_CNTF  + NUM_VOX;       // float[NUM_VOX]
static constexpr long long W_HISTF   = W_SCANV + NUM_VOX;       // float[NBUCK]
static constexpr long long W_SCANH   = W_HISTF + NBUCK;         // float[NBUCK]
static constexpr long long W_TT      = W_SCANH + NBUCK;         // float[256]
static constexpr long long W_TO      = W_TT    + 256;           // float[256]
static constexpr long long W_BVAL    = W_TO    + 256;           // float[NPTS]
static constexpr long long W_ASC     = W_BVAL  + NPTS;          // float[NPTS]

typedef __attribute__((ext_vector_type(2))) float v2f;
typedef __attribute__((ext_vector_type(8))) float v8f;

#if defined(__gfx1250__) && defined(__has_builtin)
#if __has_builtin(__builtin_amdgcn_global_load_async_to_lds_b32) && \
    __has_builtin(__builtin_amdgcn_s_wait_asynccnt)
#define HAVE_ASYNC_LDS 1
#endif
#endif
#ifndef HAVE_ASYNC_LDS
#define HAVE_ASYNC_LDS 0
#endif

// ---------------- init kernels ----------------
__global__ void k_zero_ints(int* p, long long n) {
  long long i = (long long)blockIdx.x * blockDim.x + threadIdx.x;
  if (i < n) p[i] = 0;
}

__global__ void k_init_out(float* out, const int* occ_grid, long long n) {
  long long i = (long long)blockIdx.x * blockDim.x + threadIdx.x;
  if (i >= n) return;
  float v;
  if (i < O_TABLE)      v = 0.0f;   // pts / conf / desc / desc_conf
  else if (i < O_OCC)   v = -1.0f;  // table default
  else if (i < O_IMG)   v = 0.0f;   // occupied default
  else if (i < O_GRID)  v = -1.0f;  // image id default
  else if (i < O_TOTAL) v = (float)occ_grid[i - O_GRID]; // passthrough
  else                  v = 0.0f;
  out[i] = v;
}

// ---------------- voxel id + pre-filter counts ----------------
__global__ void k_vid(const float* __restrict__ pts, const float* __restrict__ center,
                      int* __restrict__ vid, int* __restrict__ counts0) {
  int p = blockIdx.x * blockDim.x + threadIdx.x;
  if (p >= NPTS) return;
  int idx[3];
  bool inb = true;
#pragma unroll
  for (int d = 0; d < 3; ++d) {
    float mc = center[d] - HALF_CUBE;
    float q  = (pts[3 * p + d] - mc) / RESF;
    int   ii = (int)floorf(q);
    idx[d] = ii;
    inb = inb && (ii >= 0) && (ii < G);
  }
  int v = inb ? (idx[0] * (G * G) + idx[1] * G + idx[2]) : NUM_VOX;
  vid[p] = v;
  atomicAdd(&counts0[v], 1);
}

__global__ void k_countsF(const int* __restrict__ counts0, float* __restrict__ cntF,
                          float* __restrict__ out) {
  int v = blockIdx.x * blockDim.x + threadIdx.x;
  if (v >= NUM_VOX) return;
  int c = counts0[v];
  bool kept = c > MIN_PTS;
  cntF[v] = kept ? (float)c : 0.0f;
  out[O_OCC + v] = kept ? 1.0f : 0.0f;   // occupied == (table[:,0] >= 0)
}

// ---------------- confidence bucket histogram ----------------
__device__ __forceinline__ int conf_key(float c) {
  int b = (int)(c * 65536.0f);
  if (b < 0) b = 0;
  if (b > NBUCK - 1) b = NBUCK - 1;
  return b;
}

__global__ void k_hist(const float* __restrict__ conf, int* __restrict__ hist) {
  int p = blockIdx.x * blockDim.x + threadIdx.x;
  if (p >= NPTS) return;
  atomicAdd(&hist[conf_key(conf[p])], 1);
}

__global__ void k_i2f(const int* __restrict__ s, float* __restrict__ d, int n) {
  int i = blockIdx.x * blockDim.x + threadIdx.x;
  if (i < n) d[i] = (float)s[i];
}

// ---------------- WMMA tile scan ----------------
// Per 256-element tile (one wave32 block):
//   view tile as M[16][16] (row-major sequence); inclusive scan =
//   (M x U)[r][c] + sum of full rows < r,  U = upper-triangular ones.
// M x U is done with 4x V_WMMA_F32_16X16X4_F32 accumulating over K.
// All values are small integers => f32 arithmetic is exact (< 2^24).
__global__ void k_scan_tiles(const float* __restrict__ src, float* __restrict__ scan,
                             float* __restrict__ tileTot) {
  __shared__ float sm[256];
  __shared__ float rsh[16];
  const int tile = blockIdx.x;
  const int tid  = threadIdx.x;          // blockDim.x == 32 (one wave)
  const long long tb = (long long)tile * 256;
  for (int e = tid; e < 256; e += 32) sm[e] = src[tb + e];
  __syncthreads();

  const int hi = tid >> 4;       // 0/1
  const int lo = tid & 15;

  if (tid < 16) {
    float s = 0.0f;
    for (int c = 0; c < 16; ++c) s += sm[tid * 16 + c];
    rsh[tid] = s;
  }

#if defined(__gfx1250__)
  v8f acc = {0.f, 0.f, 0.f, 0.f, 0.f, 0.f, 0.f, 0.f};
#pragma unroll
  for (int kc = 0; kc < 4; ++kc) {
    const int k0 = 4 * kc + 2 * hi;
    v2f A; A.x = sm[lo * 16 + k0]; A.y = sm[lo * 16 + k0 + 1];   // A[M=lo][K]
    v2f B; B.x = (k0     <= lo) ? 1.0f : 0.0f;                   // U[K][N=lo]
           B.y = (k0 + 1 <= lo) ? 1.0f : 0.0f;
    acc = __builtin_amdgcn_wmma_f32_16x16x4_f32(
        false, A, false, B, (short)0, acc, false, false);
  }
  __syncthreads();
  {
    float run = 0.0f;
    for (int r = 0; r < 8 * hi; ++r) run += rsh[r];
#pragma unroll
    for (int v = 0; v < 8; ++v) {
      const int r = v + 8 * hi;                 // C/D layout: row = v + 8*hi
      scan[tb + r * 16 + lo] = acc[v] + run;    // col = lane & 15
      run += rsh[r];
    }
  }
#else
  // host-parse / non-gfx1250 fallback: serial tile scan
  __syncthreads();
  if (tid == 0) {
    float run = 0.0f;
    for (int e = 0; e < 256; ++e) { run += sm[e]; scan[tb + e] = run; }
  }
  __syncthreads();
#endif
  if (tid == 0) {
    float tot = 0.0f;
    for (int r = 0; r < 16; ++r) tot += rsh[r];
    tileTot[tile] = tot;
  }
}

__global__ void k_scan_tot(const float* __restrict__ tt, float* __restrict__ to, int numTiles) {
  if (threadIdx.x == 0 && blockIdx.x == 0) {
    float run = 0.0f;
    for (int t = 0; t < numTiles; ++t) { to[t] = run; run += tt[t]; }
  }
}

__global__ void k_offsets(const float* __restrict__ scan, const float* __restrict__ cnt,
                          const float* __restrict__ to, int* __restrict__ off, int n) {
  int i = blockIdx.x * blockDim.x + threadIdx.x;
  if (i >= n) return;
  off[i] = (int)(scan[i] - cnt[i] + to[i >> 8]);  // exclusive = inclusive - self + tile base
}

// ---------------- counting-sort scatters ----------------
__global__ void k_scatter_vox(const int* __restrict__ vid, const int* __restrict__ counts0,
                              const int* __restrict__ off, int* __restrict__ fill,
                              int* __restrict__ sidx) {
  int p = blockIdx.x * blockDim.x + threadIdx.x;
  if (p >= NPTS) return;
  int v = vid[p];
  if (v < NUM_VOX && counts0[v] > MIN_PTS) {
    int s = atomicAdd(&fill[v], 1);
    sidx[off[v] + s] = p;
  }
}

__global__ void k_scatter_buck(const float* __restrict__ conf, const int* __restrict__ boff,
                               int* __restrict__ bfill, float* __restrict__ bval) {
  int p = blockIdx.x * blockDim.x + threadIdx.x;
  if (p >= NPTS) return;
  float c = conf[p];
  int b = conf_key(c);
  int s = atomicAdd(&bfill[b], 1);
  bval[boff[b] + s] = c;
}

// per-bucket exact ascending value sort (k ~ 24 avg; ties are equal values)
__global__ void k_bucket_rank(const int* __restrict__ hist, const int* __restrict__ boff,
                              const float* __restrict__ bval, float* __restrict__ asc) {
  int b = blockIdx.x;
  int k = hist[b];
  if (k == 0) return;
  long long base = boff[b];
  for (int e = threadIdx.x; e < k; e += 32) {
    float v = bval[base + e];
    int r = 0;
    for (int j = 0; j < k; ++j) {
      float w = bval[base + j];
      r += (w < v) || (w == v && j < e);
    }
    asc[base + r] = v;
  }
}

// ---------------- per-voxel rank + emit ----------------
#define STAGE 512
__global__ void k_voxel_emit(const int* __restrict__ counts0, const int* __restrict__ off,
                             const int* __restrict__ sidx,
                             const float* __restrict__ pts, const float* __restrict__ conf,
                             const float* __restrict__ desc, const float* __restrict__ dconf,
                             const int* __restrict__ img, const float* __restrict__ center,
                             const float* __restrict__ asc, float* __restrict__ out) {
  __shared__ int   s_pid[STAGE];
  __shared__ float s_conf[STAGE];
  const int v = blockIdx.x;
  const int k = counts0[v];
  if (k <= MIN_PTS) return;                 // voxel filtered out (uniform per block)
  const long long base = off[v];
  const bool useLds = (k <= STAGE);

  if (useLds) {
#if HAVE_ASYNC_LDS
    // async gather: stage point ids then their confidences into LDS
    for (int e = threadIdx.x; e < k; e += blockDim.x)
      __builtin_amdgcn_global_load_async_to_lds_b32(
          (int*)(sidx + base + e), (int*)&s_pid[e], 0, 0);
    __builtin_amdgcn_s_wait_asynccnt(0);
    __syncthreads();
    for (int e = threadIdx.x; e < k; e += blockDim.x)
      __builtin_amdgcn_global_load_async_to_lds_b32(
          (int*)(conf + s_pid[e]), (int*)&s_conf[e], 0, 0);
    __builtin_amdgcn_s_wait_asynccnt(0);
    __syncthreads();
#else
    for (int e = threadIdx.x; e < k; e += blockDim.x) {
      int p = sidx[base + e];
      s_pid[e]  = p;
      s_conf[e] = conf[p];
    }
    __syncthreads();
#endif
  }

  for (int e = threadIdx.x; e < k; e += blockDim.x) {
    int   p; float c;
    if (useLds) { p = s_pid[e]; c = s_conf[e]; }
    else        { p = sidx[base + e]; c = conf[p]; }
    // rank within voxel: conf descending, tie -> original index ascending
    int r = 0;
    for (int j = 0; j < k; ++j) {
      int pj; float cj;
      if (useLds) { pj = s_pid[j]; cj = s_conf[j]; }
      else        { pj = sidx[base + j]; cj = conf[pj]; }
      r += (cj > c) || (cj == c && pj < p);
    }
    if (r < MAX_PTS) {
      const long long jo = base + r;        // global sorted position
#pragma unroll
      for (int d = 0; d < 3; ++d) {
        float mc = center[d] - HALF_CUBE;
        float pt = pts[3LL * p + d];
        float fi = floorf((pt - mc) / RESF);
        out[O_PTS + jo * 3 + d] = (pt - mc - fi * RESF - RESF * 0.5f) / (RESF * 0.5f);
      }
      out[O_CONF + jo] = asc[(long long)NPTS - 1 - p];   // conf_sorted[point_sorted] quirk
#pragma unroll 4
      for (int t = 0; t < 24; ++t)
        out[O_DESC + jo * 24 + t] = desc[24LL * p + t];
      out[O_DCONF + jo] = dconf[p];
      out[O_IMG + jo]   = (float)img[p];
      out[O_TABLE + (long long)v * MAX_PTS + r] = (float)jo;  // table stores sorted index
    }
  }
}

// ---------------- launch ----------------
extern "C" void kernel_launch(void* const* d_in, const int* in_sizes, int n_in,
                              void* d_out, int out_size, void* d_ws, size_t ws_size,
                              hipStream_t stream) {
  const float* pts    = (const float*)d_in[0];
  const float* conf   = (const float*)d_in[1];
  const float* desc   = (const float*)d_in[2];
  const float* dconf  = (const float*)d_in[3];
  const int*   img    = (const int*)d_in[4];
  const float* center = (const float*)d_in[5];
  const int*   ogrid  = (const int*)d_in[6];
  float* out = (float*)d_out;
  int*   wi  = (int*)d_ws;
  float* wf  = (float*)d_ws;

  int* counts0 = wi + W_COUNTS0;
  int* fill    = wi + W_FILL;
  int* hist    = wi + W_HIST;
  int* bfill   = wi + W_BFILL;
  int* vid     = wi + W_VID;
  int* sidx    = wi + W_SIDX;
  int* offv    = wi + W_OFF;
  int* boff    = wi + W_BOFF;
  float* cntF  = wf + W_CNTF;
  float* scanV = wf + W_SCANV;
  float* histF = wf + W_HISTF;
  float* scanH = wf + W_SCANH;
  float* tt    = wf + W_TT;
  float* to    = wf + W_TO;
  float* bval  = wf + W_BVAL;
  float* asc   = wf + W_ASC;

  const int TB = 256;
  const int gP = (NPTS + TB - 1) / TB;
  const int gV = (NUM_VOX + TB - 1) / TB;
  const int gB = (NBUCK + TB - 1) / TB;
  const long long nOut = (long long)out_size;
  const int gO = (int)((nOut + TB - 1) / TB);
  const int gZ = (int)((W_ZEND + TB - 1) / TB);

  // 0) defaults
  k_zero_ints<<<gZ, TB, 0, stream>>>(wi, W_ZEND);
  k_init_out<<<gO, TB, 0, stream>>>(out, ogrid, nOut);

  // 1) voxel ids + pre-filter counts, filtered counts, occupied
  k_vid<<<gP, TB, 0, stream>>>(pts, center, vid, counts0);
  k_countsF<<<gV, TB, 0, stream>>>(counts0, cntF, out);

  // 2) exclusive scan of filtered voxel counts (WMMA tile scan)
  k_scan_tiles<<<NUM_VOX / 256, 32, 0, stream>>>(cntF, scanV, tt);
  k_scan_tot<<<1, 32, 0, stream>>>(tt, to, NUM_VOX / 256);
  k_offsets<<<gV, TB, 0, stream>>>(scanV, cntF, to, offv, NUM_VOX);

  // 3) counting-sort points into per-voxel segments (order fixed later by rank)
  k_scatter_vox<<<gP, TB, 0, stream>>>(vid, counts0, offv, fill, sidx);

  // 4) global descending conf values (bucket sort; WMMA scan of 65536-bin histogram)
  k_hist<<<gP, TB, 0, stream>>>(conf, hist);
  k_i2f<<<gB, TB, 0, stream>>>(hist, histF, NBUCK);
  k_scan_tiles<<<NBUCK / 256, 32, 0, stream>>>(histF, scanH, tt);
  k_scan_tot<<<1, 32, 0, stream>>>(tt, to, NBUCK / 256);
  k_offsets<<<gB, TB, 0, stream>>>(scanH, histF, to, boff, NBUCK);
  k_scatter_buck<<<gP, TB, 0, stream>>>(conf, boff, bfill, bval);
  k_bucket_rank<<<NBUCK, 32, 0, stream>>>(hist, boff, bval, asc);

  // 5) per-voxel confidence ranking + emit all outputs
  k_voxel_emit<<<NUM_VOX, 64, 0, stream>>>(counts0, offv, sidx, pts, conf, desc,
                                           dconf, img, center, asc, out);
}